// SosModel_14474039788026
// MI455X (gfx1250) — compile-verified
//
#include <hip/hip_runtime.h>
#include <stdint.h>

// ---------------------------------------------------------------------------
// Cascaded biquad (4 SOS sections) over [B=16, T=32768, C=64] fp32.
// Chunked parallel scan:
//   pass1: per-(b,c,chunk) run with zero state -> particular state c_j
//   combine: per-series affine scan  s_{j+1} = A s_j + c_j,  A = M^L (8x8)
//           -> batched over 16 series per wave via V_WMMA_F32_16X16X4_F32
//   pass2: re-run chunks with correct initial states, write output
// Data movement: TDM async tensor loads (tensor_load_to_lds) double-buffered
// into LDS, waited with s_wait_tensorcnt. Memory-bound: 3*128MB ~ 16us ideal.
// ---------------------------------------------------------------------------

#define Bb 16
#define Tt 32768
#define Cc 64
#define Ss 4
#define NCHUNK 128
#define Lc (Tt / NCHUNK)   // 256 samples per chunk (power of two)
#define LOG2_LC 8
#define TSs 64             // timesteps per LDS tile
#define NT (Lc / TSs)      // 4 tiles per chunk
#define NSERIES (Bb * Cc)  // 1024 independent series
#define NSTATE 8           // 2 DF2T states x 4 sections

typedef unsigned int v4u __attribute__((ext_vector_type(4)));
typedef int          v4i __attribute__((ext_vector_type(4)));
typedef int          v8i __attribute__((ext_vector_type(8)));
typedef float        v2f __attribute__((ext_vector_type(2)));
typedef float        v8f __attribute__((ext_vector_type(8)));

#if defined(__has_builtin)
#if __has_builtin(__builtin_amdgcn_tensor_load_to_lds) && \
    __has_builtin(__builtin_amdgcn_s_wait_tensorcnt)
#define USE_TDM 1
#endif
#if __has_builtin(__builtin_amdgcn_wmma_f32_16x16x4_f32)
#define USE_WMMA 1
#endif
#endif
#ifndef USE_TDM
#define USE_TDM 0
#endif
#ifndef USE_WMMA
#define USE_WMMA 0
#endif

struct Coef {
    float b0[Ss], b1[Ss], b2[Ss], a1[Ss], a2[Ss];
};

__device__ inline void load_coefs(const float* __restrict__ sos, Coef& k) {
#pragma unroll
    for (int s = 0; s < Ss; ++s) {
        float inv = 1.0f / sos[s * 6 + 3];  // normalize by a0
        k.b0[s] = sos[s * 6 + 0] * inv;
        k.b1[s] = sos[s * 6 + 1] * inv;
        k.b2[s] = sos[s * 6 + 2] * inv;
        k.a1[s] = sos[s * 6 + 4] * inv;
        k.a2[s] = sos[s * 6 + 5] * inv;
    }
}

// One sample through the 4-section cascade, DF-II-transposed states z[8].
__device__ inline float step_sections(const Coef& k, float u, float* z) {
#pragma unroll
    for (int s = 0; s < Ss; ++s) {
        float y  = fmaf(k.b0[s], u, z[2 * s]);
        float n1 = fmaf(k.b1[s], u, z[2 * s + 1]) - k.a1[s] * y;
        float n2 = fmaf(k.b2[s], u, -k.a2[s] * y);
        z[2 * s]     = n1;
        z[2 * s + 1] = n2;
        u = y;
    }
    return u;
}

#if USE_TDM
// Issue a TDM 1D tile load: TSs*Cc contiguous fp32 (16 KB) global -> LDS.
// D# packed per CDNA5 ISA ch.8; groups 2-3 zero => <=2D tensor.
__device__ inline void tdm_issue(const float* gsrc, unsigned lds_off) {
    const unsigned NE = TSs * Cc;  // 4096 elements
    uint64_t ga = (uint64_t)(uintptr_t)gsrc;
    v4u g0;
    g0.x = 1u;                                   // count=1 (valid, user load)
    g0.y = lds_off;                              // lds_addr (bytes)
    g0.z = (unsigned)ga;                         // global_addr[31:0]
    g0.w = (unsigned)((ga >> 32) & 0x01FFFFFFu)  // global_addr[56:32]
           | (2u << 30);                         // type=2 ("image")
    v8i g1;
    g1[0] = 0x00020000;                          // data_size=2 (4 bytes)
    g1[1] = (int)((NE & 0xFFFFu) << 16);         // tensor_dim0[15:0]
    g1[2] = (int)(NE >> 16);                     // tensor_dim0[31:16], dim1=0
    g1[3] = (int)((NE & 0xFFFFu) << 16);         // tile_dim0 = NE
    g1[4] = 0;                                   // tile_dim1/2 unused
    g1[5] = (int)NE;                             // tensor_dim0_stride
    g1[6] = 0;
    g1[7] = 0;
    v4i gz4 = {0, 0, 0, 0};
    v8i gz8 = {0, 0, 0, 0, 0, 0, 0, 0};
    __builtin_amdgcn_tensor_load_to_lds(g0, g1, gz4, gz4, gz8, 0);
}
#endif

// pass1 (WRITE=false): zero init state, store final states to ws.
// pass2 (WRITE=true):  load init state from ws, write filtered output.
// Block = 64 threads (2 waves) = all channels of one (batch, chunk).
template <bool WRITE>
__global__ __launch_bounds__(64) void sos_pass(const float* __restrict__ x,
                                               const float* __restrict__ sos,
                                               float* __restrict__ out,
                                               float* __restrict__ ws) {
    __shared__ float lds[2][TSs * Cc];  // 2 x 16KB double buffer

    const int b     = blockIdx.x >> 7;           // / NCHUNK
    const int chunk = blockIdx.x & (NCHUNK - 1);
    const int c     = threadIdx.x;               // 0..63 = channel
    const int t0    = chunk * Lc;
    const float* xb = x + ((size_t)b * Tt + t0) * (size_t)Cc;

    Coef k;
    load_coefs(sos, k);

    float z[NSTATE];
    if (WRITE) {
#pragma unroll
        for (int i = 0; i < NSTATE; ++i)
            z[i] = ws[(((size_t)b * NCHUNK + chunk) * NSTATE + i) * Cc + c];
    } else {
#pragma unroll
        for (int i = 0; i < NSTATE; ++i) z[i] = 0.0f;
    }

#if USE_TDM
    const bool wave0 = (threadIdx.x < 32);
    if (wave0) tdm_issue(xb, (unsigned)(uintptr_t)&lds[0][0]);
#endif

    for (int i = 0; i < NT; ++i) {
#if USE_TDM
        if (wave0) {
            if (i + 1 < NT) {
                tdm_issue(xb + (size_t)(i + 1) * TSs * Cc,
                          (unsigned)(uintptr_t)&lds[(i + 1) & 1][0]);
                __builtin_amdgcn_s_wait_tensorcnt(1);  // tile i complete
            } else {
                __builtin_amdgcn_s_wait_tensorcnt(0);
            }
        }
        __syncthreads();
#else
        __syncthreads();
        {   // cooperative fallback copy (contiguous slab)
            const float* src = xb + (size_t)i * TSs * Cc;
            for (int e = threadIdx.x; e < TSs * Cc; e += 64)
                lds[i & 1][e] = src[e];
        }
        __syncthreads();
#endif
        const float* buf = &lds[i & 1][0];
        float* ob = WRITE ? (out + ((size_t)b * Tt + t0 + i * TSs) * Cc + c)
                          : nullptr;
#pragma unroll 4
        for (int tt = 0; tt < TSs; ++tt) {
            float u = buf[tt * Cc + c];
            float y = step_sections(k, u, z);
            if (WRITE) __builtin_nontemporal_store(y, ob + (size_t)tt * Cc);
        }
        __syncthreads();  // buffer i&1 reused by tile i+2 next iteration
    }

    if (!WRITE) {
#pragma unroll
        for (int i = 0; i < NSTATE; ++i)
            ws[(((size_t)b * NCHUNK + chunk) * NSTATE + i) * Cc + c] = z[i];
    }
}

// Affine scan over chunks. A = M^Lc (8x8) by repeated squaring; then
// s_{j+1} = A s_j + c_j, rewriting ws with each chunk's true initial state.
// WMMA path: one wave handles 16 series; S (8 states x 16 series, padded
// 16x16) is the B operand, A padded 16x16 split into two K=4 A-fragments,
// c_j is the C accumulator: D = Afrag1*S[4:8] + (Afrag0*S[0:4] + Cj).
__global__ __launch_bounds__(256) void sos_combine(const float* __restrict__ sos,
                                                   float* __restrict__ ws) {
    Coef k;
    load_coefs(sos, k);

    // Columns of M = one homogeneous step applied to basis vectors.
    float M[NSTATE][NSTATE];
    for (int j = 0; j < NSTATE; ++j) {
        float st[NSTATE];
#pragma unroll
        for (int i = 0; i < NSTATE; ++i) st[i] = (i == j) ? 1.0f : 0.0f;
        (void)step_sections(k, 0.0f, st);
#pragma unroll
        for (int i = 0; i < NSTATE; ++i) M[i][j] = st[i];
    }
    // A = M^(2^LOG2_LC)
    float A[NSTATE][NSTATE], Tm[NSTATE][NSTATE];
    for (int i = 0; i < NSTATE; ++i)
        for (int j = 0; j < NSTATE; ++j) A[i][j] = M[i][j];
    for (int q = 0; q < LOG2_LC; ++q) {
        for (int i = 0; i < NSTATE; ++i)
            for (int j = 0; j < NSTATE; ++j) {
                float acc = 0.0f;
#pragma unroll
                for (int kk = 0; kk < NSTATE; ++kk)
                    acc = fmaf(A[i][kk], A[kk][j], acc);
                Tm[i][j] = acc;
            }
        for (int i = 0; i < NSTATE; ++i)
            for (int j = 0; j < NSTATE; ++j) A[i][j] = Tm[i][j];
    }

#if USE_WMMA
    // One wave per 16 series. 64 waves total (grid = 8 blocks x 256 thr).
    const int lane = threadIdx.x & 31;
    const int wave = (blockIdx.x * blockDim.x + threadIdx.x) >> 5;  // 0..63
    const int b    = wave >> 2;
    const int c0   = (wave & 3) << 4;   // first of 16 series (channels)
    const int m    = lane & 15;
    const bool hi  = lane >= 16;

    // A-fragments (16x4 layout: VGPR0=K0 lanes0-15 / K2 lanes16-31, VGPR1=K1/K3)
    // Fragment q covers padded-A columns 4q..4q+3; rows/cols >=8 are zero.
    v2f af0, af1;
    {
        int j0 = hi ? 2 : 0;
        af0.x = (m < 8 && (j0 + 0) < 8) ? A[m][j0 + 0] : 0.0f;
        af0.y = (m < 8 && (j0 + 1) < 8) ? A[m][j0 + 1] : 0.0f;
        int j1 = 4 + j0;
        af1.x = (m < 8 && (j1 + 0) < 8) ? A[m][j1 + 0] : 0.0f;
        af1.y = (m < 8 && (j1 + 1) < 8) ? A[m][j1 + 1] : 0.0f;
    }

    // State fragments in B layout (4x16 per fragment):
    // s0: rows 0-3 (VGPR0 = row0 lanes<16 / row2 lanes>=16; VGPR1 = row1/row3)
    // s1: rows 4-7 likewise.
    v2f s0 = {0.0f, 0.0f}, s1 = {0.0f, 0.0f};

    for (int ch = 0; ch < NCHUNK; ++ch) {
        const size_t base = ((size_t)(b * NCHUNK + ch) * NSTATE) * Cc + c0;

        // c_j in C/D layout: VGPR r = row r (lanes<16); rows 8-15 zero.
        v8f cf;
#pragma unroll
        for (int r = 0; r < NSTATE; ++r)
            cf[r] = hi ? 0.0f : ws[base + (size_t)r * Cc + m];

        // Store current state (the chunk's true initial state) back to ws.
        if (!hi) {
            ws[base + 0 * Cc + m] = s0.x;
            ws[base + 1 * Cc + m] = s0.y;
            ws[base + 4 * Cc + m] = s1.x;
            ws[base + 5 * Cc + m] = s1.y;
        } else {
            ws[base + 2 * Cc + m] = s0.x;
            ws[base + 3 * Cc + m] = s0.y;
            ws[base + 6 * Cc + m] = s1.x;
            ws[base + 7 * Cc + m] = s1.y;
        }

        // D = A[:,4:8]*S[4:8,:] + (A[:,0:4]*S[0:4,:] + Cj)
        v8f d = __builtin_amdgcn_wmma_f32_16x16x4_f32(
            false, af0, false, s0, (short)0, cf, false, false);
        d = __builtin_amdgcn_wmma_f32_16x16x4_f32(
            false, af1, false, s1, (short)0, d, false, false);

        // Convert D layout (row r in VGPR r, lanes<16) back to B layout.
        float u0 = __shfl(d[0], m, 32), u2 = __shfl(d[2], m, 32);
        float u1 = __shfl(d[1], m, 32), u3 = __shfl(d[3], m, 32);
        float u4 = __shfl(d[4], m, 32), u6 = __shfl(d[6], m, 32);
        float u5 = __shfl(d[5], m, 32), u7 = __shfl(d[7], m, 32);
        s0.x = hi ? u2 : u0;
        s0.y = hi ? u3 : u1;
        s1.x = hi ? u6 : u4;
        s1.y = hi ? u7 : u5;
    }
#else
    // Scalar fallback: one thread per series.
    int tid = blockIdx.x * blockDim.x + threadIdx.x;
    if (tid >= NSERIES) return;
    int b = tid >> 6;
    int c = tid & 63;
    float s[NSTATE];
#pragma unroll
    for (int i = 0; i < NSTATE; ++i) s[i] = 0.0f;
    for (int ch = 0; ch < NCHUNK; ++ch) {
        size_t base = (((size_t)b * NCHUNK + ch) * NSTATE) * Cc + c;
        float cj[NSTATE], ns[NSTATE];
#pragma unroll
        for (int i = 0; i < NSTATE; ++i) cj[i] = ws[base + (size_t)i * Cc];
#pragma unroll
        for (int i = 0; i < NSTATE; ++i) ws[base + (size_t)i * Cc] = s[i];
        for (int i = 0; i < NSTATE; ++i) {
            float acc = cj[i];
#pragma unroll
            for (int kk = 0; kk < NSTATE; ++kk)
                acc = fmaf(A[i][kk], s[kk], acc);
            ns[i] = acc;
        }
#pragma unroll
        for (int i = 0; i < NSTATE; ++i) s[i] = ns[i];
    }
#endif
}

extern "C" void kernel_launch(void* const* d_in, const int* in_sizes, int n_in,
                              void* d_out, int out_size, void* d_ws,
                              size_t ws_size, hipStream_t stream) {
    (void)in_sizes; (void)n_in; (void)out_size; (void)ws_size;
    const float* x   = (const float*)d_in[0];  // [B,T,C] fp32
    const float* sos = (const float*)d_in[1];  // [4,6] fp32
    float* out = (float*)d_out;                // [B,T,C] fp32
    float* ws  = (float*)d_ws;                 // needs B*NCHUNK*8*C*4 = 4 MB

    dim3 grid(Bb * NCHUNK);  // 2048 blocks
    dim3 blk(64);
    sos_pass<false><<<grid, blk, 0, stream>>>(x, sos, out, ws);
    // 8 blocks x 256 threads = 64 waves (WMMA: one wave per 16 series).
    sos_combine<<<dim3(8), dim3(256), 0, stream>>>(sos, ws);
    sos_pass<true><<<grid, blk, 0, stream>>>(x, sos, out, ws);
}